// GCN_26783416058429
// MI455X (gfx1250) — compile-verified
//
#include <hip/hip_runtime.h>
#include <hip/hip_bf16.h>

typedef __attribute__((ext_vector_type(2))) float v2f;
typedef __attribute__((ext_vector_type(8))) float v8f;

#ifndef __has_builtin
#define __has_builtin(x) 0
#endif
#if __has_builtin(__builtin_amdgcn_wmma_f32_16x16x4_f32)
#define HAVE_WMMA_F32 1
#else
#define HAVE_WMMA_F32 0
#endif

#define NEG_SLOPE 0.25f

// Native fp32 global atomic add (GLOBAL_ATOMIC_ADD_F32, no-return form).
// unsafeAtomicAdd carries the no-fine-grained-memory guarantee so the backend
// selects the hardware atomic instead of a CAS retry loop.
__device__ __forceinline__ void gatomic_fadd(float* p, float v) {
  (void)unsafeAtomicAdd(p, v);
}

// ---------------------------------------------------------------------------
// Dense GEMM: S[N, dout] = H[N, DIN] @ W[DIN, dout]
// One wave32 per 16x16 output tile, V_WMMA_F32_16X16X4_F32, K=4 per step.
// Fragment layouts per CDNA5 ISA 7.12.2:
//   A (16x4 f32):  lanes 0-15 rows M=0..15 with K={0,1} in VGPR0/1,
//                  lanes 16-31 same rows with K={2,3}
//   B (4x16 f32):  lanes 0-15 cols N=0..15 with K={0,1}, lanes 16-31 K={2,3}
//   C/D (16x16):   VGPR v -> row v (lanes 0-15) / row v+8 (lanes 16-31)
// ---------------------------------------------------------------------------
template <int DIN>
__global__ __launch_bounds__(256) void gcn_gemm_wmma(
    const float* __restrict__ H, const float* __restrict__ W,
    float* __restrict__ S, int N, int dout) {
  const int lane   = threadIdx.x & 31;
  const int waveId = blockIdx.x * (blockDim.x >> 5) + (threadIdx.x >> 5);
  const int tiles_n = dout >> 4;
  const int tile_m  = waveId / tiles_n;
  const int m0      = tile_m << 4;
  if (m0 >= N) return;  // wave-uniform exit; EXEC stays all-1 for WMMA
  const int n0  = (waveId - tile_m * tiles_n) << 4;
  const int l15 = lane & 15;
  const int khi = lane >> 4;  // 0 for lanes 0-15, 1 for lanes 16-31

  int rowA = m0 + l15;
  if (rowA > N - 1) rowA = N - 1;  // clamp load (keeps EXEC full); store guarded
  const float* __restrict__ Ap = H + (size_t)rowA * DIN + (khi << 1);
  const float* __restrict__ Bp = W + (size_t)(khi << 1) * dout + n0 + l15;

  v8f c = {};
#if HAVE_WMMA_F32
#pragma unroll
  for (int k = 0; k < DIN; k += 4) {
    v2f a, b;
    a.x = Ap[k];                       // A[row][k + 2*khi]
    a.y = Ap[k + 1];                   // A[row][k + 2*khi + 1]
    b.x = Bp[(size_t)k * dout];        // W[k + 2*khi    ][col]
    b.y = Bp[(size_t)k * dout + dout]; // W[k + 2*khi + 1][col]
    c = __builtin_amdgcn_wmma_f32_16x16x4_f32(false, a, false, b, (short)0, c,
                                              false, false);
  }
#else
  // Scalar fallback producing the identical C-fragment layout.
  {
    const int colF = n0 + l15;
#pragma unroll
    for (int v = 0; v < 8; ++v) {
      int row = m0 + (khi << 3) + v;
      if (row > N - 1) row = N - 1;
      float acc = 0.f;
      for (int k = 0; k < DIN; ++k)
        acc += H[(size_t)row * DIN + k] * W[(size_t)k * dout + colF];
      c[v] = acc;
    }
  }
#endif

  const int col   = n0 + l15;
  const int rbase = m0 + (khi << 3);
#pragma unroll
  for (int v = 0; v < 8; ++v) {
    const int row = rbase + v;
    if (row < N) S[(size_t)row * dout + col] = c[v];
  }
}

// Generic (non-16-multiple) fallback GEMM: one thread per output element.
__global__ void gcn_gemm_generic(const float* __restrict__ H,
                                 const float* __restrict__ W,
                                 float* __restrict__ S, int N, int din,
                                 int dout) {
  long long i = (long long)blockIdx.x * blockDim.x + threadIdx.x;
  if (i >= (long long)N * dout) return;
  const int r = (int)(i / dout), cix = (int)(i % dout);
  float acc = 0.f;
  for (int k = 0; k < din; ++k)
    acc += H[(size_t)r * din + k] * W[(size_t)k * dout + cix];
  S[i] = acc;
}

__global__ void zero_f32(float* __restrict__ p, long long n) {
  long long i = (long long)blockIdx.x * blockDim.x + threadIdx.x;
  if (i < n) p[i] = 0.f;
}

// ---------------------------------------------------------------------------
// Edge scatter: A[row] += w * S[col], 4 floats per thread, (d/4) threads/edge.
// 16 consecutive lanes cover one support row -> fully coalesced float4 gather.
// Support + accumulator buffers are L2-resident (<< 192 MB), so the random
// per-edge access pattern hits L2, not HBM. Scatter uses native (no-return)
// GLOBAL_ATOMIC_ADD_F32 at the L2 atomic units.
// ---------------------------------------------------------------------------
__global__ void gcn_scatter(const int* __restrict__ erow,
                            const int* __restrict__ ecol,
                            const float* __restrict__ ew,
                            const float* __restrict__ S, float* __restrict__ A,
                            int E, int d, int shift) {
  const long long idx = (long long)blockIdx.x * blockDim.x + threadIdx.x;
  const long long e = idx >> shift;
  if (e >= E) return;
  const int j = (int)(idx & ((1 << shift) - 1));
  const int r = erow[e];
  const int cix = ecol[e];
  const float w = ew[e];
  const float4 s = *((const float4*)(S + (size_t)cix * d) + j);
  float* dst = A + (size_t)r * d + ((size_t)j << 2);
  gatomic_fadd(dst + 0, w * s.x);
  gatomic_fadd(dst + 1, w * s.y);
  gatomic_fadd(dst + 2, w * s.z);
  gatomic_fadd(dst + 3, w * s.w);
}

// Fused bias + leaky_relu(0.25), in place (accumulator becomes next input).
__global__ void gcn_bias_lrelu(float* __restrict__ A, const float* __restrict__ b,
                               long long n_total, int d) {
  long long i = (long long)blockIdx.x * blockDim.x + threadIdx.x;
  if (i >= n_total) return;
  const int col = (int)(i % d);
  const float v = A[i] + b[col];
  A[i] = v >= 0.f ? v : NEG_SLOPE * v;
}

// ---------------------------------------------------------------------------
extern "C" void kernel_launch(void* const* d_in, const int* in_sizes, int n_in,
                              void* d_out, int out_size, void* d_ws,
                              size_t ws_size, hipStream_t stream) {
  const float* x   = (const float*)d_in[0];
  const int*   er  = (const int*)d_in[1];
  const int*   ec  = (const int*)d_in[2];
  const float* ew  = (const float*)d_in[3];
  const float* W1  = (const float*)d_in[4];
  const float* b1  = (const float*)d_in[5];
  const float* W2  = (const float*)d_in[6];
  const float* b2  = (const float*)d_in[7];
  const float* W3  = (const float*)d_in[8];
  const float* b3  = (const float*)d_in[9];

  const int E  = in_sizes[1];
  const int d1 = in_sizes[5];
  const int d2 = in_sizes[7];
  const int d3 = in_sizes[9];
  const int d0 = in_sizes[4] / d1;   // 64
  const int N  = in_sizes[0] / d0;   // 100000

  float* ws   = (float*)d_ws;
  float* sup1 = ws;                         // N*d1
  float* agg1 = sup1 + (size_t)N * d1;      // N*d1
  float* sup2 = agg1 + (size_t)N * d1;      // N*d2
  float* agg2 = sup2 + (size_t)N * d2;      // N*d2
  float* sup3 = agg2 + (size_t)N * d2;      // N*d3
  float* agg3 = (float*)d_out;              // N*d3 -> final output

  auto cdiv = [](long long a, long long b) { return (int)((a + b - 1) / b); };

  auto run_layer = [&](const float* Hin, int din, const float* W,
                       const float* b, int dout, float* sup, float* agg) {
    // 1) dense GEMM (WMMA path for the known layer shapes)
    if ((dout & 15) == 0 && din == 64) {
      const int tiles = cdiv(N, 16) * (dout >> 4);
      gcn_gemm_wmma<64><<<cdiv((long long)tiles * 32, 256), 256, 0, stream>>>(
          Hin, W, sup, N, dout);
    } else if ((dout & 15) == 0 && din == 32) {
      const int tiles = cdiv(N, 16) * (dout >> 4);
      gcn_gemm_wmma<32><<<cdiv((long long)tiles * 32, 256), 256, 0, stream>>>(
          Hin, W, sup, N, dout);
    } else {
      const long long t = (long long)N * dout;
      gcn_gemm_generic<<<cdiv(t, 256), 256, 0, stream>>>(Hin, W, sup, N, din,
                                                         dout);
    }
    // 2) zero accumulator (must re-init every call; harness doesn't)
    const long long nd = (long long)N * dout;
    zero_f32<<<cdiv(nd, 256), 256, 0, stream>>>(agg, nd);
    // 3) edge gather/scale/scatter-add (native f32 atomics)
    const int shift = __builtin_ctz(dout >> 2);  // threads/edge = d/4
    const long long tot = (long long)E << shift;
    gcn_scatter<<<cdiv(tot, 256), 256, 0, stream>>>(er, ec, ew, sup, agg, E,
                                                    dout, shift);
    // 4) bias + leaky_relu in place
    gcn_bias_lrelu<<<cdiv(nd, 256), 256, 0, stream>>>(agg, b, nd, dout);
  };

  run_layer(x,    d0, W1, b1, d1, sup1, agg1);
  run_layer(agg1, d1, W2, b2, d2, sup2, agg2);
  run_layer(agg2, d2, W3, b3, d3, sup3, agg3);
}